// SparseVanillaAttentionLayer_35399120454219
// MI455X (gfx1250) — compile-verified
//
#include <hip/hip_runtime.h>
#include <math.h>

#define NN 100000       // nodes
#define NE 1600000      // edges
#define HID 128
#define NH  4
#define HD  32

typedef float v2f __attribute__((ext_vector_type(2)));
typedef float v8f __attribute__((ext_vector_type(8)));

// ---------------------------------------------------------------------------
// fp32 WMMA GEMM:  out[M,Ncols] = A[M,K] @ B[K,Ncols] + bias[Ncols]
// One wave32 computes a 16 x (TN*16) strip via V_WMMA_F32_16X16X4_F32, with
// TN independent accumulator chains for ILP and 1/TN the A-operand traffic.
// Grid guarantees waves == (M/16) * (Ncols/16/TN); K%4==0, all compile-time.
// ---------------------------------------------------------------------------
template <int K, int Ncols, int TN>
__global__ void wmma_gemm_f32(const float* __restrict__ A,
                              const float* __restrict__ B,
                              const float* __restrict__ bias,
                              float* __restrict__ out) {
  const int wave = (int)((blockIdx.x * blockDim.x + threadIdx.x) >> 5);
  const int lane = (int)(threadIdx.x & 31);
  constexpr int groups = (Ncols >> 4) / TN;   // N-tile groups per M row
  const int tm = wave / groups;
  const int tg = wave - tm * groups;
  const int m0 = tm << 4;
  const int n0 = tg * (TN * 16);

  const int laneM = lane & 15;   // M index (A) / N index (B,C)
  const int kHi   = lane >> 4;   // 0: K={0,1}, C rows m0+r ; 1: K={2,3}, m0+8+r

  v8f acc[TN];
#pragma unroll
  for (int t = 0; t < TN; ++t) acc[t] = (v8f){};

  const float* arow  = A + (size_t)(m0 + laneM) * K + (kHi << 1);
  const float* bbase = B + (size_t)(kHi << 1) * Ncols + n0 + laneM;

#pragma unroll
  for (int kk = 0; kk < K; kk += 4) {
    v2f a;
    a.x = arow[0];
    a.y = arow[1];
#pragma unroll
    for (int t = 0; t < TN; ++t) {
      v2f b;
      b.x = bbase[t * 16];
      b.y = bbase[Ncols + t * 16];
      acc[t] = __builtin_amdgcn_wmma_f32_16x16x4_f32(false, a, false, b,
                                                     (short)0, acc[t],
                                                     false, false);
    }
    arow  += 4;
    bbase += (size_t)4 * Ncols;
  }

  const int mBase = m0 + (kHi << 3);
#pragma unroll
  for (int t = 0; t < TN; ++t) {
    const float bv = bias[n0 + t * 16 + laneM];
#pragma unroll
    for (int r = 0; r < 8; ++r) {
      out[(size_t)(mBase + r) * Ncols + n0 + t * 16 + laneM] = acc[t][r] + bv;
    }
  }
}

// ---------------------------------------------------------------------------
// init: m = -inf, s = 0, z = 0  (must run every call; graph replays don't
// re-poison and we must not carry state across calls)
// ---------------------------------------------------------------------------
__global__ void init_buffers(float* __restrict__ m, float* __restrict__ s,
                             float* __restrict__ z, int nMS, int nZ) {
  int t = blockIdx.x * blockDim.x + threadIdx.x;
  if (t < nMS) { m[t] = -INFINITY; s[t] = 0.0f; }
  if (t < nZ)  { z[t] = 0.0f; }
}

// float atomic-max via signed/unsigned ordering trick (works with -inf init)
__device__ __forceinline__ void atomicMaxF32(float* addr, float val) {
  if (val >= 0.0f) {
    atomicMax((int*)addr, __float_as_int(val));
  } else {
    atomicMin((unsigned int*)addr, (unsigned int)__float_as_int(val));
  }
}

// ---------------------------------------------------------------------------
// One wave per edge: per-head dot(Q[row], K[col]) * scale, store score,
// atomic segment-max into m[row][head].
// qkv layout per node: [0,128)=Q  [128,256)=K  [256,384)=V
// ---------------------------------------------------------------------------
__global__ void edge_scores(const float* __restrict__ qkv,
                            const long long* __restrict__ edges,
                            float* __restrict__ scores,
                            float* __restrict__ mbuf) {
  const int wave = (int)((blockIdx.x * blockDim.x + threadIdx.x) >> 5);
  const int lane = (int)(threadIdx.x & 31);
  if (wave >= NE) return;                       // wave-uniform
  const int row = (int)edges[wave];
  const int col = (int)edges[(size_t)NE + wave];
  const int d0   = lane << 2;                   // 4 contiguous dims per lane
  const int head = lane >> 3;                   // 8 lanes per head

  const float4 q = *(const float4*)(qkv + (size_t)row * 384 + d0);
  const float4 k = *(const float4*)(qkv + (size_t)col * 384 + 128 + d0);
  float p = q.x * k.x + q.y * k.y + q.z * k.z + q.w * k.w;
  p += __shfl_xor(p, 1);
  p += __shfl_xor(p, 2);
  p += __shfl_xor(p, 4);                        // 8-lane head reduction

  if ((lane & 7) == 0) {
    const float sc = p * 0.17677669529663689f;  // 1/sqrt(32)
    scores[(size_t)wave * NH + head] = sc;
    atomicMaxF32(mbuf + (size_t)row * NH + head, sc);
  }
}

// ---------------------------------------------------------------------------
// Per (edge, head): e = exp(score - m[row]); scores <- e; s[row] += e
// ---------------------------------------------------------------------------
__global__ void edge_expsum(const long long* __restrict__ edges,
                            const float* __restrict__ mbuf,
                            float* __restrict__ scores,
                            float* __restrict__ sbuf) {
  int t = blockIdx.x * blockDim.x + threadIdx.x;
  if (t >= NE * NH) return;
  int e = t >> 2;
  int h = t & 3;
  int row = (int)edges[e];
  float ev = __expf(scores[t] - mbuf[(size_t)row * NH + h]);
  scores[t] = ev;
  atomicAdd(sbuf + (size_t)row * NH + h, ev);
}

// ---------------------------------------------------------------------------
// Per (edge, dim): z[row][d] += (e/s[row][h]) * V[col][d]  (coalesced in d)
// ---------------------------------------------------------------------------
__global__ void edge_scatter(const float* __restrict__ qkv,
                             const long long* __restrict__ edges,
                             const float* __restrict__ scores,
                             const float* __restrict__ sbuf,
                             float* __restrict__ z) {
  long long t = (long long)blockIdx.x * blockDim.x + threadIdx.x;
  if (t >= (long long)NE * HID) return;
  int e = (int)(t >> 7);
  int d = (int)(t & 127);
  int h = d >> 5;
  int row = (int)edges[e];
  int col = (int)edges[(size_t)NE + e];
  float attn = scores[(size_t)e * NH + h] / sbuf[(size_t)row * NH + h];
  float v = qkv[(size_t)col * 384 + 256 + d];
  atomicAdd(z + (size_t)row * HID + d, attn * v);
}

// ---------------------------------------------------------------------------
extern "C" void kernel_launch(void* const* d_in, const int* in_sizes, int n_in,
                              void* d_out, int out_size, void* d_ws, size_t ws_size,
                              hipStream_t stream) {
  const float*     x      = (const float*)d_in[0];
  const long long* edges  = (const long long*)d_in[1];   // int64 [2, NE]
  const float*     w_qkv  = (const float*)d_in[2];
  const float*     b_qkv  = (const float*)d_in[3];
  const float*     w_o    = (const float*)d_in[4];
  const float*     b_o    = (const float*)d_in[5];
  float*           out    = (float*)d_out;

  // scratch layout (floats)
  float* ws     = (float*)d_ws;
  float* qkv    = ws;                               // NN*384
  float* scores = qkv    + (size_t)NN * 384;        // NE*NH
  float* mbuf   = scores + (size_t)NE * NH;         // NN*NH
  float* sbuf   = mbuf   + (size_t)NN * NH;         // NN*NH
  float* z      = sbuf   + (size_t)NN * NH;         // NN*HID

  const int nZ  = NN * HID;   // 12.8M
  const int nMS = NN * NH;    // 400k
  init_buffers<<<(nZ + 255) / 256, 256, 0, stream>>>(mbuf, sbuf, z, nMS, nZ);

  // QKV GEMM: (100000/16)*(384/64) = 37500 waves -> 9375 blocks x 4 waves
  wmma_gemm_f32<HID, 384, 4><<<9375, 128, 0, stream>>>(x, w_qkv, b_qkv, qkv);

  // 1 wave per edge: 1.6M waves -> 200000 blocks x 8 waves
  edge_scores<<<NE / 8, 256, 0, stream>>>(qkv, edges, scores, mbuf);

  edge_expsum<<<(NE * NH + 255) / 256, 256, 0, stream>>>(edges, mbuf, scores, sbuf);

  edge_scatter<<<(int)(((long long)NE * HID + 255) / 256), 256, 0, stream>>>(
      qkv, edges, scores, sbuf, z);

  // Output GEMM: (100000/16)*(128/64) = 12500 waves -> 3125 blocks x 4 waves
  wmma_gemm_f32<HID, HID, 4><<<3125, 128, 0, stream>>>(z, w_o, b_o, out);
}